// PointSDF_81389630259789
// MI455X (gfx1250) — compile-verified
//
#include <hip/hip_runtime.h>
#include <math.h>

typedef __attribute__((ext_vector_type(2))) float v2f;
typedef __attribute__((ext_vector_type(8))) float v8f;

#define BB 8
#define NN 4096
#define SS 512
#define KK 32
#define MROWS (BB*SS*KK)   // 131072

// ---------------------------------------------------------------------------
// 1) density: inv_dens[b,i] = (N*2.5*BW) / sum_j exp(-d2_ij * 2*BW^2)
// ---------------------------------------------------------------------------
__global__ void density_kernel(const float* __restrict__ xyz, float* __restrict__ inv_dens) {
    __shared__ float tx[256], ty[256], tz[256];
    int b = blockIdx.x >> 4;
    int i = (blockIdx.x & 15) * 256 + threadIdx.x;
    const float* xb = xyz + (size_t)b * 3 * NN;
    float xi = xb[i], yi = xb[NN + i], zi = xb[2 * NN + i];
    float sum = 0.f;
    for (int jt = 0; jt < NN; jt += 256) {
        int j = jt + threadIdx.x;
        tx[threadIdx.x] = xb[j];
        ty[threadIdx.x] = xb[NN + j];
        tz[threadIdx.x] = xb[2 * NN + j];
        __syncthreads();
        #pragma unroll 8
        for (int jj = 0; jj < 256; ++jj) {
            float dx = xi - tx[jj], dy = yi - ty[jj], dz = zi - tz[jj];
            float d2 = dx * dx + dy * dy + dz * dz;
            sum += __expf(-d2 * 0.02f);
        }
        __syncthreads();
    }
    inv_dens[b * NN + i] = (NN * 0.25f) / sum;
}

// ---------------------------------------------------------------------------
// 2) farthest point sampling; writes new_xyz transposed [B,3,S] directly to out
// ---------------------------------------------------------------------------
__global__ __launch_bounds__(1024) void fps_kernel(const float* __restrict__ xyz,
                                                   float* __restrict__ out_newxyz) {
    __shared__ float lx[NN], ly[NN], lz[NN];
    __shared__ float rv[1024];
    __shared__ int   ri[1024];
    __shared__ int   s_far;
    int b = blockIdx.x, tid = threadIdx.x;
    const float* xb = xyz + (size_t)b * 3 * NN;
    for (int j = tid; j < NN; j += 1024) { lx[j] = xb[j]; ly[j] = xb[NN + j]; lz[j] = xb[2 * NN + j]; }
    float dist[4];
    #pragma unroll
    for (int t = 0; t < 4; ++t) dist[t] = 1e10f;
    if (tid == 0) s_far = 0;
    __syncthreads();
    for (int it = 0; it < SS; ++it) {
        int far = s_far;
        if (tid == 0) {
            out_newxyz[b * 3 * SS + 0 * SS + it] = lx[far];
            out_newxyz[b * 3 * SS + 1 * SS + it] = ly[far];
            out_newxyz[b * 3 * SS + 2 * SS + it] = lz[far];
        }
        float cx = lx[far], cy = ly[far], cz = lz[far];
        float bv = -1.f; int bi = 0;
        #pragma unroll
        for (int t = 0; t < 4; ++t) {
            int j = tid + t * 1024;
            float dx = lx[j] - cx, dy = ly[j] - cy, dz = lz[j] - cz;
            float d = dx * dx + dy * dy + dz * dz;
            dist[t] = fminf(dist[t], d);
            if (dist[t] > bv) { bv = dist[t]; bi = j; }  // strict > keeps lowest idx on tie
        }
        rv[tid] = bv; ri[tid] = bi;
        __syncthreads();
        for (int off = 512; off > 0; off >>= 1) {
            if (tid < off) {
                float ov = rv[tid + off]; int oi = ri[tid + off];
                if (ov > rv[tid] || (ov == rv[tid] && oi < ri[tid])) { rv[tid] = ov; ri[tid] = oi; }
            }
            __syncthreads();
        }
        if (tid == 0) s_far = ri[0];
        __syncthreads();
    }
}

// ---------------------------------------------------------------------------
// 3) kNN (32 smallest d2, ties -> lower index, ascending order)
// ---------------------------------------------------------------------------
__global__ void knn_kernel(const float* __restrict__ xyz, const float* __restrict__ newxyz,
                           int* __restrict__ knn) {
    __shared__ float d2s[NN];
    __shared__ float rv[128];
    __shared__ int   ri[128];
    int bs = blockIdx.x; int b = bs >> 9; int s = bs & 511;
    int tid = threadIdx.x;
    const float* xb = xyz + (size_t)b * 3 * NN;
    float cx = newxyz[b * 3 * SS + s];
    float cy = newxyz[b * 3 * SS + SS + s];
    float cz = newxyz[b * 3 * SS + 2 * SS + s];
    for (int j = tid; j < NN; j += 128) {
        float dx = xb[j] - cx, dy = xb[NN + j] - cy, dz = xb[2 * NN + j] - cz;
        d2s[j] = dx * dx + dy * dy + dz * dz;
    }
    __syncthreads();
    for (int it = 0; it < KK; ++it) {
        float bvv = 3.4e38f; int bii = 0;
        for (int j = tid; j < NN; j += 128) {
            float v = d2s[j];
            if (v < bvv) { bvv = v; bii = j; }
        }
        rv[tid] = bvv; ri[tid] = bii;
        __syncthreads();
        for (int off = 64; off > 0; off >>= 1) {
            if (tid < off) {
                float ov = rv[tid + off]; int oi = ri[tid + off];
                if (ov < rv[tid] || (ov == rv[tid] && oi < ri[tid])) { rv[tid] = ov; ri[tid] = oi; }
            }
            __syncthreads();
        }
        if (tid == 0) { knn[bs * KK + it] = ri[0]; d2s[ri[0]] = 3.4e38f; }
        __syncthreads();
    }
}

// ---------------------------------------------------------------------------
// 4) gather + build MLP inputs
// ---------------------------------------------------------------------------
__global__ void build_kernel(const float* __restrict__ xyz, const float* __restrict__ points,
                             const float* __restrict__ newxyz, const int* __restrict__ knn,
                             const float* __restrict__ inv_dens,
                             float* __restrict__ A0, float* __restrict__ Aw, float* __restrict__ Ad) {
    __shared__ float gd[KK];
    int bs = blockIdx.x; int b = bs >> 9; int s = bs & 511;
    int k = threadIdx.x;
    int idx = knn[bs * KK + k];
    const float* xb = xyz + (size_t)b * 3 * NN;
    const float* fb = points + (size_t)b * 3 * NN;
    float cx = newxyz[b * 3 * SS + s];
    float cy = newxyz[b * 3 * SS + SS + s];
    float cz = newxyz[b * 3 * SS + 2 * SS + s];
    float gx = xb[idx] - cx, gy = xb[NN + idx] - cy, gz = xb[2 * NN + idx] - cz;
    int m = bs * KK + k;
    A0[m * 6 + 0] = gx; A0[m * 6 + 1] = gy; A0[m * 6 + 2] = gz;
    A0[m * 6 + 3] = fb[idx]; A0[m * 6 + 4] = fb[NN + idx]; A0[m * 6 + 5] = fb[2 * NN + idx];
    Aw[m * 3 + 0] = gx; Aw[m * 3 + 1] = gy; Aw[m * 3 + 2] = gz;
    float gdv = inv_dens[b * NN + idx];
    gd[k] = gdv;
    __syncthreads();
    float mx = gd[0];
    for (int t = 1; t < KK; ++t) mx = fmaxf(mx, gd[t]);
    Ad[m] = gdv / mx;
}

// ---------------------------------------------------------------------------
// 5a) fast WMMA GEMM: requires Cin%4==0, Cout%32==0, M%32==0.
//     One wave computes a 32x32 output tile (2x2 WMMA accumulators):
//     per k-step 4 vector loads feed 4 v_wmma_f32_16x16x4_f32.
// ---------------------------------------------------------------------------
__global__ void gemm_wmma_2x2(const float* __restrict__ A, const float* __restrict__ W,
                              const float* __restrict__ bias, float* __restrict__ Y,
                              int M, int Cin, int Cout) {
    int lane = threadIdx.x & 31;
    int half = lane >> 4;           // selects K pair {0,1} vs {2,3}
    int l = lane & 15;
    int wpb = blockDim.x >> 5;
    int gw = blockIdx.x * wpb + (threadIdx.x >> 5);
    int nwaves = gridDim.x * wpb;
    int mtiles = M >> 5;
    int ntiles = Cout >> 5;
    int tiles = mtiles * ntiles;
    for (int t = gw; t < tiles; t += nwaves) {
        int mt = t % mtiles, nt = t / mtiles;
        int m0 = mt << 5, n0 = nt << 5;
        const float* A0p = A + (size_t)(m0 + l) * Cin + (half << 1);
        const float* A1p = A0p + (size_t)16 * Cin;
        const float* W0p = W + (size_t)(n0 + l) * Cin + (half << 1);
        const float* W1p = W0p + (size_t)16 * Cin;
        v8f c00 = {}, c01 = {}, c10 = {}, c11 = {};
        #pragma unroll 4
        for (int k0 = 0; k0 < Cin; k0 += 4) {
            __builtin_prefetch(A0p + k0 + 256, 0, 0);
            __builtin_prefetch(W0p + k0 + 256, 0, 0);
            v2f a0 = *(const v2f*)(A0p + k0);   // 8B aligned: Cin%4==0, even offsets
            v2f a1 = *(const v2f*)(A1p + k0);
            v2f b0 = *(const v2f*)(W0p + k0);
            v2f b1 = *(const v2f*)(W1p + k0);
            c00 = __builtin_amdgcn_wmma_f32_16x16x4_f32(false, a0, false, b0, (short)0, c00, false, false);
            c01 = __builtin_amdgcn_wmma_f32_16x16x4_f32(false, a0, false, b1, (short)0, c01, false, false);
            c10 = __builtin_amdgcn_wmma_f32_16x16x4_f32(false, a1, false, b0, (short)0, c10, false, false);
            c11 = __builtin_amdgcn_wmma_f32_16x16x4_f32(false, a1, false, b1, (short)0, c11, false, false);
        }
        float bc0 = bias[n0 + l];
        float bc1 = bias[n0 + 16 + l];
        #pragma unroll
        for (int r = 0; r < 8; ++r) {
            int row0 = m0 + r + (half << 3);
            int row1 = row0 + 16;
            Y[(size_t)row0 * Cout + n0 + l]      = c00[r] + bc0;
            Y[(size_t)row0 * Cout + n0 + 16 + l] = c01[r] + bc1;
            Y[(size_t)row1 * Cout + n0 + l]      = c10[r] + bc0;
            Y[(size_t)row1 * Cout + n0 + 16 + l] = c11[r] + bc1;
        }
    }
}

// ---------------------------------------------------------------------------
// 5b) generic WMMA GEMM for ragged shapes: branchless guards (clamped addresses
//     + value selects) so no EXEC-divergent loads appear in the inner loop.
// ---------------------------------------------------------------------------
__global__ void gemm_wmma_gen(const float* __restrict__ A, const float* __restrict__ W,
                              const float* __restrict__ bias, float* __restrict__ Y,
                              int M, int Cin, int Cout) {
    int lane = threadIdx.x & 31;
    int half = lane >> 4;
    int l = lane & 15;
    int wpb = blockDim.x >> 5;
    int gw = blockIdx.x * wpb + (threadIdx.x >> 5);
    int nwaves = gridDim.x * wpb;
    int mtiles = M >> 4;
    int ntiles = (Cout + 15) >> 4;
    int tiles = mtiles * ntiles;
    for (int t = gw; t < tiles; t += nwaves) {
        int mt = t % mtiles, nt = t / mtiles;
        int m0 = mt << 4, n0 = nt << 4;
        int row = m0 + l;
        int col = n0 + l;
        bool colok = col < Cout;
        int colc = colok ? col : n0;            // n0 always < Cout
        const float* Ap = A + (size_t)row * Cin;
        const float* Wp = W + (size_t)colc * Cin;
        v8f c = {};
        for (int k0 = 0; k0 < Cin; k0 += 4) {
            int ka = k0 + (half << 1);
            bool k0ok = ka < Cin;
            bool k1ok = (ka + 1) < Cin;
            int ka0 = k0ok ? ka : 0;            // clamped -> unconditional load
            int ka1 = k1ok ? (ka + 1) : 0;
            float a0 = Ap[ka0]; a0 = k0ok ? a0 : 0.f;
            float a1 = Ap[ka1]; a1 = k1ok ? a1 : 0.f;
            float b0 = Wp[ka0]; b0 = (colok && k0ok) ? b0 : 0.f;
            float b1 = Wp[ka1]; b1 = (colok && k1ok) ? b1 : 0.f;
            v2f av = {a0, a1};
            v2f bv = {b0, b1};
            c = __builtin_amdgcn_wmma_f32_16x16x4_f32(false, av, false, bv, (short)0, c,
                                                      false, false);
        }
        float bcol = bias[colc];
        #pragma unroll
        for (int r = 0; r < 8; ++r) {
            int orow = m0 + r + (half << 3);
            if (colok) Y[(size_t)orow * Cout + col] = c[r] + bcol;
        }
    }
}

// ---------------------------------------------------------------------------
// 6) per-(b,s) einsum via WMMA: fused[bs, c*16+w] = sum_k feat[bs,k,c]*d[bs,k]*w[bs,k,w]
// ---------------------------------------------------------------------------
__global__ void einsum_wmma(const float* __restrict__ feat, const float* __restrict__ wbuf,
                            const float* __restrict__ dscale, float* __restrict__ fused) {
    int lane = threadIdx.x & 31;
    int half = lane >> 4, l = lane & 15;
    int gw = blockIdx.x * (blockDim.x >> 5) + (threadIdx.x >> 5);  // 0..32767
    int bs = gw >> 3; int mt = gw & 7;
    int m0 = mt << 4;
    const float* fb = feat + (size_t)bs * KK * 128;
    const float* wb = wbuf + (size_t)bs * KK * 16;
    const float* db = dscale + (size_t)bs * KK;
    v8f c = {};
    #pragma unroll
    for (int k0 = 0; k0 < KK; k0 += 4) {
        int ka = k0 + (half << 1);
        float a0 = fb[ka * 128 + m0 + l];
        float a1 = fb[(ka + 1) * 128 + m0 + l];
        float b0 = db[ka]     * wb[ka * 16 + l];
        float b1 = db[ka + 1] * wb[(ka + 1) * 16 + l];
        v2f av = {a0, a1};
        v2f bv = {b0, b1};
        c = __builtin_amdgcn_wmma_f32_16x16x4_f32(false, av, false, bv, (short)0, c,
                                                  false, false);
    }
    float* ob = fused + (size_t)bs * 2048;
    #pragma unroll
    for (int r = 0; r < 8; ++r) ob[(m0 + r + (half << 3)) * 16 + l] = c[r];
}

// ---------------------------------------------------------------------------
// 7) training-mode BN: per-channel mean/var (block per channel, f64 accumulation)
// ---------------------------------------------------------------------------
__global__ void bn_stats(const float* __restrict__ Y, float* __restrict__ meanvar,
                         int Mrows, int C) {
    __shared__ double s1[256], s2[256];
    int c = blockIdx.x, tid = threadIdx.x;
    double a1 = 0.0, a2 = 0.0;
    for (int r = tid; r < Mrows; r += 256) {
        double v = (double)Y[(size_t)r * C + c];
        a1 += v; a2 += v * v;
    }
    s1[tid] = a1; s2[tid] = a2;
    __syncthreads();
    for (int off = 128; off > 0; off >>= 1) {
        if (tid < off) { s1[tid] += s1[tid + off]; s2[tid] += s2[tid + off]; }
        __syncthreads();
    }
    if (tid == 0) {
        double mu = s1[0] / Mrows;
        double var = s2[0] / Mrows - mu * mu;
        meanvar[c] = (float)mu;
        meanvar[C + c] = (float)var;
    }
}

__global__ void bn_apply(const float* __restrict__ Yin, float* __restrict__ Yout,
                         const float* __restrict__ meanvar, const float* __restrict__ gamma,
                         const float* __restrict__ beta, int Mrows, int C) {
    int total = Mrows * C;
    for (int i = blockIdx.x * blockDim.x + threadIdx.x; i < total; i += gridDim.x * blockDim.x) {
        int c = i % C;
        float mu = meanvar[c], var = meanvar[C + c];
        float z = gamma[c] * (Yin[i] - mu) * rsqrtf(var + 1e-5f) + beta[c];
        Yout[i] = fmaxf(z, 0.f);
    }
}

__global__ void bn_apply_final(const float* __restrict__ Yin, float* __restrict__ outp,
                               const float* __restrict__ meanvar, const float* __restrict__ gamma,
                               const float* __restrict__ beta) {
    int i = blockIdx.x * blockDim.x + threadIdx.x;   // over 4096*128
    if (i >= 4096 * 128) return;
    int c = i & 127; int bs = i >> 7;
    int b = bs >> 9; int s = bs & 511;
    float mu = meanvar[c], var = meanvar[128 + c];
    float z = gamma[c] * (Yin[i] - mu) * rsqrtf(var + 1e-5f) + beta[c];
    outp[b * 65536 + c * 512 + s] = fmaxf(z, 0.f);
}

// ---------------------------------------------------------------------------
// host driver
// ---------------------------------------------------------------------------
static inline void run_gemm(const float* A, const float* W, const float* bias, float* Y,
                            int M, int Cin, int Cout, hipStream_t stream) {
    if ((Cin % 4 == 0) && (Cout % 32 == 0) && (M % 32 == 0)) {
        int tiles = (M >> 5) * (Cout >> 5);
        int blocks = (tiles + 7) / 8;
        gemm_wmma_2x2<<<blocks, 256, 0, stream>>>(A, W, bias, Y, M, Cin, Cout);
    } else {
        int tiles = (M >> 4) * ((Cout + 15) >> 4);
        int blocks = (tiles + 7) / 8;
        gemm_wmma_gen<<<blocks, 256, 0, stream>>>(A, W, bias, Y, M, Cin, Cout);
    }
}

static inline void run_bn(const float* Yin, float* Yout, float* mv,
                          const float* g, const float* be, int M, int C, hipStream_t stream) {
    bn_stats<<<C, 256, 0, stream>>>(Yin, mv, M, C);
    int total = M * C;
    int blocks = (total + 255) / 256;
    if (blocks > 16384) blocks = 16384;
    bn_apply<<<blocks, 256, 0, stream>>>(Yin, Yout, mv, g, be, M, C);
}

extern "C" void kernel_launch(void* const* d_in, const int* in_sizes, int n_in,
                              void* d_out, int out_size, void* d_ws, size_t ws_size,
                              hipStream_t stream) {
    (void)in_sizes; (void)n_in; (void)out_size; (void)ws_size;
    const float* xyz    = (const float*)d_in[0];
    const float* points = (const float*)d_in[1];
    const float* mW[3]  = {(const float*)d_in[2],  (const float*)d_in[6],  (const float*)d_in[10]};
    const float* mb[3]  = {(const float*)d_in[3],  (const float*)d_in[7],  (const float*)d_in[11]};
    const float* mg[3]  = {(const float*)d_in[4],  (const float*)d_in[8],  (const float*)d_in[12]};
    const float* mbe[3] = {(const float*)d_in[5],  (const float*)d_in[9],  (const float*)d_in[13]};
    const float* wW[3]  = {(const float*)d_in[14], (const float*)d_in[18], (const float*)d_in[22]};
    const float* wb[3]  = {(const float*)d_in[15], (const float*)d_in[19], (const float*)d_in[23]};
    const float* wg[3]  = {(const float*)d_in[16], (const float*)d_in[20], (const float*)d_in[24]};
    const float* wbe[3] = {(const float*)d_in[17], (const float*)d_in[21], (const float*)d_in[25]};
    const float* dW[3]  = {(const float*)d_in[26], (const float*)d_in[30], (const float*)d_in[34]};
    const float* db_[3] = {(const float*)d_in[27], (const float*)d_in[31], (const float*)d_in[35]};
    const float* dg[3]  = {(const float*)d_in[28], (const float*)d_in[32], (const float*)d_in[36]};
    const float* dbe[3] = {(const float*)d_in[29], (const float*)d_in[33], (const float*)d_in[37]};
    const float* linW = (const float*)d_in[38];
    const float* linb = (const float*)d_in[39];
    const float* blg  = (const float*)d_in[40];
    const float* blb  = (const float*)d_in[41];

    float* out_newxyz = (float*)d_out;                  // [B,3,S]
    float* out_feat   = (float*)d_out + BB * 3 * SS;    // [B,128,S]

    float* ws = (float*)d_ws;
    size_t o = 0;
    float* inv_dens = ws + o; o += (size_t)BB * NN;
    int*   knn      = (int*)(ws + o); o += (size_t)MROWS;
    float* A0       = ws + o; o += (size_t)MROWS * 6;
    float* Aw       = ws + o; o += (size_t)MROWS * 3;
    float* Ad       = ws + o; o += (size_t)MROWS;
    float* h0       = ws + o; o += (size_t)MROWS * 64;
    float* h1       = ws + o; o += (size_t)MROWS * 64;
    float* feat     = ws + o; o += (size_t)MROWS * 128;
    float* olin     = ws + o; o += (size_t)4096 * 128;
    float* mv       = ws + o; o += 256;
    // lifetime reuse: after mlp chain, h0/h1 are free
    float* w0buf = h0;
    float* w1buf = h0 + (size_t)MROWS * 8;
    float* w2buf = h0 + (size_t)MROWS * 16;
    float* d0buf = h0 + (size_t)MROWS * 32;
    float* d1buf = h0 + (size_t)MROWS * 48;
    float* dsbuf = h0 + (size_t)MROWS * 56;
    float* fused = h1;                                   // M*64 == 4096*2048 exactly

    density_kernel<<<BB * 16, 256, 0, stream>>>(xyz, inv_dens);
    fps_kernel<<<BB, 1024, 0, stream>>>(xyz, out_newxyz);
    knn_kernel<<<BB * SS, 128, 0, stream>>>(xyz, out_newxyz, knn);
    build_kernel<<<BB * SS, KK, 0, stream>>>(xyz, points, out_newxyz, knn, inv_dens, A0, Aw, Ad);

    // feature MLP 6 -> 64 -> 64 -> 128
    run_gemm(A0, mW[0], mb[0], h0, MROWS, 6, 64, stream);
    run_bn(h0, h0, mv, mg[0], mbe[0], MROWS, 64, stream);
    run_gemm(h0, mW[1], mb[1], h1, MROWS, 64, 64, stream);
    run_bn(h1, h1, mv, mg[1], mbe[1], MROWS, 64, stream);
    run_gemm(h1, mW[2], mb[2], feat, MROWS, 64, 128, stream);
    run_bn(feat, feat, mv, mg[2], mbe[2], MROWS, 128, stream);

    // weight net 3 -> 8 -> 8 -> 16
    run_gemm(Aw, wW[0], wb[0], w0buf, MROWS, 3, 8, stream);
    run_bn(w0buf, w0buf, mv, wg[0], wbe[0], MROWS, 8, stream);
    run_gemm(w0buf, wW[1], wb[1], w1buf, MROWS, 8, 8, stream);
    run_bn(w1buf, w1buf, mv, wg[1], wbe[1], MROWS, 8, stream);
    run_gemm(w1buf, wW[2], wb[2], w2buf, MROWS, 8, 16, stream);
    run_bn(w2buf, w2buf, mv, wg[2], wbe[2], MROWS, 16, stream);

    // density net 1 -> 16 -> 8 -> 1
    run_gemm(Ad, dW[0], db_[0], d0buf, MROWS, 1, 16, stream);
    run_bn(d0buf, d0buf, mv, dg[0], dbe[0], MROWS, 16, stream);
    run_gemm(d0buf, dW[1], db_[1], d1buf, MROWS, 16, 8, stream);
    run_bn(d1buf, d1buf, mv, dg[1], dbe[1], MROWS, 8, stream);
    run_gemm(d1buf, dW[2], db_[2], dsbuf, MROWS, 8, 1, stream);
    run_bn(dsbuf, dsbuf, mv, dg[2], dbe[2], MROWS, 1, stream);

    // einsum -> fused [4096, 2048]
    einsum_wmma<<<BB * SS, 256, 0, stream>>>(feat, w2buf, dsbuf, fused);

    // final linear 2048 -> 128 + BN + ReLU (transposed store)
    run_gemm(fused, linW, linb, olin, BB * SS, 2048, 128, stream);
    bn_stats<<<128, 256, 0, stream>>>(olin, mv, BB * SS, 128);
    bn_apply_final<<<(4096 * 128 + 255) / 256, 256, 0, stream>>>(olin, out_feat, mv, blg, blb);
}